// kernel_generated_0_21887153341378
// MI455X (gfx1250) — compile-verified
//
#include <hip/hip_runtime.h>
#include <hip/hip_bf16.h>

typedef __attribute__((ext_vector_type(2))) float v2f;
typedef __attribute__((ext_vector_type(8))) float v8f;

// Grouped conv (2 groups of 64 in-ch -> 16 out-ch, k=3 along W, zero-pad after
// circular +1 roll) as two 16x192x3136 GEMMs on V_WMMA_F32_16X16X4_F32.
//
// K ordering: K = j*64 + i (j = tap, i = in-channel within group); A and B use
// the same ordering so the contraction is unchanged, but j is constant across
// each 64-wide K segment -> patch geometry hoisted out of the hot loop and all
// hot-loop addresses are base + compile-time immediate offset.
//
// W is shared by both groups, so one wave computes BOTH groups for its 16
// spatial positions with two accumulators sharing every A fragment:
// 3 loads per WMMA instead of 4, and two independent WMMA dependency chains.
//
// 196 waves total (one per 16-wide spatial tile), 96 WMMAs per wave.

__global__ __launch_bounds__(128) void grouped_conv_wmma_f32(
    const float* __restrict__ x,   // (128, 56, 56)
    const float* __restrict__ W,   // (64, 16, 3) laid out i*48 + k*3 + j
    float* __restrict__ y) {       // (32, 56, 56)
  const int lane = threadIdx.x & 31;
  const int wave = threadIdx.x >> 5;
  const int tile = blockIdx.x * 4 + wave;      // 0..195 spatial tiles

  const int half = lane >> 4;                  // K sub-block selector (0/1)
  const int lid  = lane & 15;                  // M for A-frag, N for B-frag

  // Spatial position owned by this lane's B/D column.
  const int s = tile * 16 + lid;
  const int n = s / 56;                        // input H row
  const int m = s % 56;                        // output W col

  v8f acc0 = {};                               // group o=0 (channels 0..63)
  v8f acc1 = {};                               // group o=1 (channels 64..127)

  #pragma unroll
  for (int j = 0; j < 3; ++j) {
    // patches[c, n, j, m] = t[c, n, m+j-1] if in range else 0,
    // with t[c,n,w] = x[c,n,(w-1) mod 56]  (roll +1 before zero-pad).
    const int w_src = m + j - 1;
    const bool valid = (w_src >= 0) && (w_src < 56);
    int xw = w_src - 1;
    if (xw < 0) xw += 56;                      // wrap of the pre-roll
    if (!valid) xw = 0;                        // any in-bounds addr; masked
    const float fj = valid ? 1.0f : 0.0f;

    const float* xb = x + n * 56 + xw;         // + c*3136 per channel
    const float* Wb = W + lid * 3 + j;         // + i*48 per channel

    #pragma unroll
    for (int t = 0; t < 16; ++t) {
      const int i0 = 4 * t + 2 * half;         // this lane's first K (channel)

      // A fragment (shared by both groups): A[M=lid, K] = W[i, lid, j]
      v2f a;
      a.x = Wb[i0 * 48];
      a.y = Wb[i0 * 48 + 48];

      // B fragments: masked patches for channels i0, i0+1 (group 0)
      // and i0+64, i0+65 (group 1) -- same base, immediate offsets.
      v2f b0, b1;
      b0.x = xb[i0 * 3136] * fj;
      b0.y = xb[(i0 + 1) * 3136] * fj;
      b1.x = xb[(i0 + 64) * 3136] * fj;
      b1.y = xb[(i0 + 65) * 3136] * fj;

      acc0 = __builtin_amdgcn_wmma_f32_16x16x4_f32(
          false, a, false, b0, (short)0, acc0, false, false);
      acc1 = __builtin_amdgcn_wmma_f32_16x16x4_f32(
          false, a, false, b1, (short)0, acc1, false, false);
    }
  }

  // D layout: VGPR r, lane -> row M = r + 8*half (out-channel k), col N = lid.
  // Final roll(+1, H): row computed at n is stored at (n+1) % 56.
  const int n_out = (n + 1 == 56) ? 0 : n + 1;
  float* yo = y + n_out * 56 + m;

  #pragma unroll
  for (int r = 0; r < 8; ++r) {
    const int k = r + 8 * half;
    yo[k * 3136] = acc0[r];                    // oc = k       (group 0)
    yo[(k + 16) * 3136] = acc1[r];             // oc = 16 + k  (group 1)
  }
}

extern "C" void kernel_launch(void* const* d_in, const int* in_sizes, int n_in,
                              void* d_out, int out_size, void* d_ws, size_t ws_size,
                              hipStream_t stream) {
  const float* x = (const float*)d_in[0];   // (1,128,56,56) f32
  const float* W = (const float*)d_in[1];   // (64,16,3)     f32
  float* y = (float*)d_out;                 // (1,32,56,56)  f32

  // 196 spatial tiles, 4 waves (128 threads) per block -> 49 blocks, exact.
  grouped_conv_wmma_f32<<<49, 128, 0, stream>>>(x, W, y);
}